// GCN_50551765074154
// MI455X (gfx1250) — compile-verified
//
#include <hip/hip_runtime.h>
#include <hip/hip_bf16.h>

// GCN 2-layer, but reference returns only log_softmax(H2)[0] (node 0, 40 logits).
// Strategy: backward-slice the graph. Only deg[] needs a full edge pass; the
// two GCN layers are evaluated on the 1-hop / 2-hop in-neighborhoods of node 0
// (~33 / ~1100 nodes expected). Edge list (51 MB int32) is L2-resident (192 MB)
// after the first pass, so the 4 edge scans cost ~one HBM read total.
// The only real GEMM (X[S2,:] @ W1, 512->16) runs on V_WMMA_F32_16X16X4_F32
// (full f32 precision, matches reference numerics).

#define F_IN  512
#define HID   16
#define NCLS  40
#define CAP   32768   // max compacted 2-hop frontier rows (expected ~1100)

typedef __attribute__((ext_vector_type(2))) float v2f;
typedef __attribute__((ext_vector_type(8))) float v8f;

// ---------------- init: zero deg/flags/agg/m2, set flag1[0]=1, cnt=0 ----------
__global__ void k_init(int* __restrict__ deg, int* __restrict__ f1, int* __restrict__ f2,
                       int* __restrict__ cnt, float* __restrict__ agg,
                       float* __restrict__ m2, int N) {
    long i = (long)blockIdx.x * blockDim.x + threadIdx.x;
    long tot = (long)N * HID;
    if (i >= tot) return;
    agg[i] = 0.0f;
    if (i < N) { deg[i] = 0; f2[i] = 0; f1[i] = (i == 0) ? 1 : 0; }
    if (i < HID) m2[i] = 0.0f;
    if (i == 0) *cnt = 0;
}

// ------------- edge pass 1: degree count + mark in-neighbors of node 0 -------
__global__ void k_deg_flag1(const int* __restrict__ esrc, const int* __restrict__ edst,
                            int* __restrict__ deg, int* __restrict__ f1, int E) {
    int e = blockIdx.x * blockDim.x + threadIdx.x;
    if (e >= E) return;
    __builtin_prefetch(edst + e + 4096, 0, 1);   // global_prefetch, streaming scan
    int d = edst[e];
    atomicAdd(&deg[d], 1);
    if (d == 0) f1[esrc[e]] = 1;
}

// ---------------- dinv = rsqrt(deg + 1)  (self-loop included) ----------------
__global__ void k_dinv(const int* __restrict__ deg, float* __restrict__ dinv, int N) {
    int i = blockIdx.x * blockDim.x + threadIdx.x;
    if (i >= N) return;
    dinv[i] = rsqrtf((float)(deg[i] + 1));
}

// ------------- edge pass 2: mark 2-hop frontier (src of edges into S1) -------
__global__ void k_flag2(const int* __restrict__ esrc, const int* __restrict__ edst,
                        const int* __restrict__ f1, int* __restrict__ f2, int E) {
    int e = blockIdx.x * blockDim.x + threadIdx.x;
    if (e >= E) return;
    int d = edst[e];
    if (f1[d]) f2[esrc[e]] = 1;
}

// ---------------- compact S1 ∪ S2 into a dense row list ----------------------
__global__ void k_compact(const int* __restrict__ f1, const int* __restrict__ f2,
                          int* __restrict__ list, int* __restrict__ cnt, int N) {
    int i = blockIdx.x * blockDim.x + threadIdx.x;
    if (i >= N) return;
    if (f1[i] | f2[i]) {
        int p = atomicAdd(cnt, 1);
        if (p < CAP) list[p] = i;
    }
}

// -------- WMMA GEMM: xw[row,:] = X[row,:] @ W1 for gathered rows -------------
// One wave per 16-row tile. f32 16x16x4 WMMA, K-loop over 512.
// A layout (ISA 7.12.2): lanes 0-15 hold M=lane, K={k0,k0+1}; lanes 16-31 K={k0+2,k0+3}.
// B (4x16): lanes 0-15 hold N=lane, K={k0,k0+1}; lanes 16-31 K={k0+2,k0+3}.
// C/D: lane holds col N=lane&15; VGPR r holds row r + 8*(lane>>4).
__global__ void __launch_bounds__(32)
k_gemm(const float* __restrict__ X, const float* __restrict__ W1,
       const int* __restrict__ list, const int* __restrict__ cnt,
       float* __restrict__ xw) {
    int c = *cnt;
    if (c > CAP) c = CAP;
    int tile = blockIdx.x;
    if (tile * 16 >= c) return;           // wave-uniform exit, EXEC stays all-1s

    int lane = threadIdx.x;
    int m    = lane & 15;                 // row (A) / col (B,C)
    int half = lane >> 4;                 // 0 or 1
    int kh   = half * 2;                  // K sub-offset within group of 4

    int idx  = tile * 16 + m;
    int ridx = (idx < c) ? idx : (c - 1); // clamp gather for partial tiles
    int row  = list[ridx];
    const float* xr = X + (long)row * F_IN;

    v8f acc = {0.f, 0.f, 0.f, 0.f, 0.f, 0.f, 0.f, 0.f};
#pragma unroll 4
    for (int k0 = 0; k0 < F_IN; k0 += 4) {
        v2f a = *(const v2f*)(xr + k0 + kh);          // 8B-aligned b64 load
        v2f b;
        b.x = W1[(k0 + kh)     * HID + m];
        b.y = W1[(k0 + kh + 1) * HID + m];
        acc = __builtin_amdgcn_wmma_f32_16x16x4_f32(
            /*neg_a=*/false, a, /*neg_b=*/false, b,
            /*c_mod=*/(short)0, acc, /*reuse_a=*/false, /*reuse_b=*/false);
    }
    // predicated stores only after the last WMMA
    for (int r = 0; r < 8; ++r) {
        int mm = r + 8 * half;
        int gi = tile * 16 + mm;
        if (gi < c) {
            int orow = list[gi];
            xw[(long)orow * HID + m] = acc[r];
        }
    }
}

// ------- edge pass 3: layer-1 aggregation into S1 rows (agg buffer) ----------
__global__ void k_agg(const int* __restrict__ esrc, const int* __restrict__ edst,
                      const int* __restrict__ f1, const float* __restrict__ dinv,
                      const float* __restrict__ xw, float* __restrict__ agg, int E) {
    int e = blockIdx.x * blockDim.x + threadIdx.x;
    if (e >= E) return;
    int d = edst[e];
    if (!f1[d]) return;                   // ~1000 of 3.2M edges survive
    int s = esrc[e];
    float w = dinv[s] * dinv[d];
    const float* xs = xw + (long)s * HID;
    float* ad = agg + (long)d * HID;
#pragma unroll
    for (int ch = 0; ch < HID; ++ch) atomicAdd(&ad[ch], w * xs[ch]);
}

// ------- h1 = relu(agg + dinv^2 * xw_self + b1)  (in place on agg) -----------
__global__ void k_relu(const int* __restrict__ f1, const float* __restrict__ dinv,
                       const float* __restrict__ xw, const float* __restrict__ b1,
                       float* __restrict__ agg, int N) {
    int i = blockIdx.x * blockDim.x + threadIdx.x;
    if (i >= N) return;
    if (!f1[i]) return;
    float di = dinv[i];
    float w = di * di;
#pragma unroll
    for (int ch = 0; ch < HID; ++ch) {
        long o = (long)i * HID + ch;
        float v = agg[o] + w * xw[o] + b1[ch];
        agg[o] = fmaxf(v, 0.0f);
    }
}

// ------- edge pass 4: pool dinv[s]*h1[s] over edges into node 0 --------------
__global__ void k_pool(const int* __restrict__ esrc, const int* __restrict__ edst,
                       const float* __restrict__ dinv, const float* __restrict__ h1,
                       float* __restrict__ m2, int E) {
    int e = blockIdx.x * blockDim.x + threadIdx.x;
    if (e >= E) return;
    if (edst[e] != 0) return;
    int s = esrc[e];
    float w = dinv[s];
    const float* hs = h1 + (long)s * HID;
#pragma unroll
    for (int ch = 0; ch < HID; ++ch) atomicAdd(&m2[ch], w * hs[ch]);
}

// ------- final: out = log_softmax(dinv0 * ((m2 + dinv0*h1[0]) @ W2) + b2) ----
__global__ void k_final(const float* __restrict__ dinv, const float* __restrict__ h1,
                        const float* __restrict__ m2, const float* __restrict__ W2,
                        const float* __restrict__ b2, float* __restrict__ out) {
    __shared__ float mf[HID];
    __shared__ float t[NCLS];
    __shared__ float red[2];
    int tid = threadIdx.x;
    float d0 = dinv[0];
    if (tid < HID) mf[tid] = m2[tid] + d0 * h1[tid];   // self-loop of node 0
    __syncthreads();
    if (tid < NCLS) {
        float a = 0.0f;
#pragma unroll
        for (int ch = 0; ch < HID; ++ch) a += mf[ch] * W2[ch * NCLS + tid];
        t[tid] = d0 * a + b2[tid];
    }
    __syncthreads();
    if (tid == 0) {
        float mx = t[0];
        for (int j = 1; j < NCLS; ++j) mx = fmaxf(mx, t[j]);
        float s = 0.0f;
        for (int j = 0; j < NCLS; ++j) s += expf(t[j] - mx);
        red[0] = mx; red[1] = logf(s);
    }
    __syncthreads();
    if (tid < NCLS) out[tid] = t[tid] - red[0] - red[1];
}

extern "C" void kernel_launch(void* const* d_in, const int* in_sizes, int n_in,
                              void* d_out, int out_size, void* d_ws, size_t ws_size,
                              hipStream_t stream) {
    (void)n_in; (void)out_size; (void)ws_size;
    const int*   ei = (const int*)d_in[0];      // edge_index [2, E]
    const float* X  = (const float*)d_in[1];    // [N, 512]
    const float* W1 = (const float*)d_in[2];    // [512, 16]
    const float* b1 = (const float*)d_in[3];    // [16]
    const float* W2 = (const float*)d_in[4];    // [16, 40]
    const float* b2 = (const float*)d_in[5];    // [40]
    float* out = (float*)d_out;                 // [40]

    int E = in_sizes[0] / 2;
    int N = in_sizes[1] / F_IN;
    const int* esrc = ei;
    const int* edst = ei + E;

    // Workspace carve (~14.6 MB total)
    char* p = (char*)d_ws;
    int*   deg  = (int*)p;    p += (size_t)N * 4;
    float* dinv = (float*)p;  p += (size_t)N * 4;
    int*   f1   = (int*)p;    p += (size_t)N * 4;
    int*   f2   = (int*)p;    p += (size_t)N * 4;
    int*   list = (int*)p;    p += (size_t)CAP * 4;
    int*   cnt  = (int*)p;    p += 64;
    float* xw   = (float*)p;  p += (size_t)N * HID * 4;   // X @ W1 (sliced rows)
    float* agg  = (float*)p;  p += (size_t)N * HID * 4;   // layer-1 agg -> h1
    float* m2   = (float*)p;  p += 64;                    // node-0 pooled vector

    int gInit = (int)(((long)N * HID + 255) / 256);
    int gE    = (E + 255) / 256;
    int gN    = (N + 255) / 256;

    k_init     <<<gInit, 256, 0, stream>>>(deg, f1, f2, cnt, agg, m2, N);
    k_deg_flag1<<<gE,    256, 0, stream>>>(esrc, edst, deg, f1, E);
    k_dinv     <<<gN,    256, 0, stream>>>(deg, dinv, N);
    k_flag2    <<<gE,    256, 0, stream>>>(esrc, edst, f1, f2, E);
    k_compact  <<<gN,    256, 0, stream>>>(f1, f2, list, cnt, N);
    k_gemm     <<<CAP/16, 32, 0, stream>>>(X, W1, list, cnt, xw);
    k_agg      <<<gE,    256, 0, stream>>>(esrc, edst, f1, dinv, xw, agg, E);
    k_relu     <<<gN,    256, 0, stream>>>(f1, dinv, xw, b1, agg, N);
    k_pool     <<<gE,    256, 0, stream>>>(esrc, edst, dinv, agg, m2, E);
    k_final    <<<1,      64, 0, stream>>>(dinv, agg, m2, W2, b2, out);
}